// Layer_2_3_stim_32280974197139
// MI455X (gfx1250) — compile-verified
//
#include <hip/hip_runtime.h>

// CDNA5 / gfx1250 — HH neuron dynamics with dense Laplacian coupling.
// Heavy part: batched GEMV adj(128x1024x1024) @ V fused with rowsum(adj) via
// V_WMMA_F32_16X16X4_F32: B-matrix = (V | ones | 0...), so one accumulator
// carries adj·V (col 0) and rowsum (col 1). Memory-bound: 537 MB streamed
// once -> ~23 us floor at 23.3 TB/s. This version maximizes MLP: 8 loads in
// flight per wave (unroll 8) + 4-way K-split (32768 waves) + prefetch.

typedef __attribute__((ext_vector_type(2))) float v2f;
typedef __attribute__((ext_vector_type(8))) float v8f;

#define B_NEURONS 128
#define N_NODES   1024
#define KSPLIT    4            // K-quarters per tile, one wave each
#define KPW       (N_NODES / KSPLIT)   // 256 K per wave
#define ROWS_PER_BLOCK 32      // 2 tiles of 16 rows
#define PF_DIST   128          // prefetch 512 B ahead per row stream

__global__ __launch_bounds__(256) void hh_wmma_kernel(
    const float* __restrict__ V,     const float* __restrict__ m,
    const float* __restrict__ h,     const float* __restrict__ n,
    const float* __restrict__ I_inj, const float* __restrict__ adj,
    const float* __restrict__ g_Na,  const float* __restrict__ g_K,
    const float* __restrict__ g_leak,const float* __restrict__ E_Na,
    const float* __restrict__ E_K,   const float* __restrict__ E_leak,
    const float* __restrict__ Cm,    const float* __restrict__ V_shift,
    float* __restrict__ out)
{
    __shared__ float Vsh[N_NODES];           // V[b, :] broadcast copy
    __shared__ float red[2][KSPLIT][2][16];  // [tile][kq][dot|sum][row]

    const int tid  = threadIdx.x;
    const int wave = tid >> 5;
    const int lane = tid & 31;
    const int b    = blockIdx.x >> 5;        // 32 blocks per batch
    const int grp  = blockIdx.x & 31;        // 32-row group within batch
    const int tile = wave >> 2;              // 0..1 : which 16-row tile
    const int kq   = wave & 3;               // 0..3 : which K-quarter

    // Stage V[b,:] into LDS (coalesced, uniform trip count -> no divergence).
    for (int i = tid; i < N_NODES; i += 256)
        Vsh[i] = V[(size_t)b * N_NODES + i];
    __syncthreads();

    // A-matrix (16x4 f32) lane roles: lanes 0-15 -> K=0,1 ; lanes 16-31 -> K=2,3
    // B-matrix (4x16 f32) lane roles: col = lane&15 ; same K split by lane>>4
    const int   arow = lane & 15;            // A row index == B column index
    const int   half = lane >> 4;            // K-half selector
    const float selV = (arow == 0) ? 1.0f : 0.0f;  // B col 0 carries V
    const float sel1 = (arow == 1) ? 1.0f : 0.0f;  // B col 1 carries ones
    const int   n0   = grp * ROWS_PER_BLOCK + tile * 16;
    const float* rowbase =
        adj + ((size_t)b * N_NODES + (size_t)(n0 + arow)) * N_NODES + kq * KPW;
    const float* rowptr = rowbase + 2 * half;
    const int   vbase = kq * KPW + 2 * half;

    v8f acc = {0.f, 0.f, 0.f, 0.f, 0.f, 0.f, 0.f, 0.f};

    for (int kc = 0; kc < KPW; kc += 32) {
        // Speculative streaming prefetch: row-aligned so halves dedupe.
        __builtin_prefetch(rowbase + kc + PF_DIST, 0, 1);
        #pragma unroll
        for (int u = 0; u < 8; ++u) {
            const int k0 = kc + u * 4;
            // 16 rows streamed sequentially, 16B per row per K=4 step.
            v2f a = *(const v2f*)(rowptr + k0);
            // Uniform LDS broadcast of V, routed into B cols 0/1 via selects.
            float vk0 = Vsh[vbase + k0];
            float vk1 = Vsh[vbase + k0 + 1];
            v2f bb;
            bb.x = fmaf(vk0, selV, sel1);
            bb.y = fmaf(vk1, selV, sel1);
            // D = A x B + C : col0 += adj·V, col1 += rowsum.
            acc = __builtin_amdgcn_wmma_f32_16x16x4_f32(
                /*neg_a=*/false, a, /*neg_b=*/false, bb,
                /*c_mod=*/(short)0, acc, /*reuse_a=*/false, /*reuse_b=*/false);
        }
    }

    // C/D layout: VGPR r holds M=r (lanes 0-15, N=lane) and M=8+r (lanes 16-31).
    // Columns 0 and 1 live in lanes {0,16} and {1,17}; spill via LDS.
    if (arow < 2) {
        const int rb = half * 8;
        #pragma unroll
        for (int r = 0; r < 8; ++r)
            red[tile][kq][arow][rb + r] = acc[r];
    }
    __syncthreads();

    // Epilogue: threads 0..31 finish one node each (pointwise HH part).
    if (tid < 32) {
        const int et = tid >> 4;             // tile
        const int er = tid & 15;             // row within tile
        const float dotV = red[et][0][0][er] + red[et][1][0][er]
                         + red[et][2][0][er] + red[et][3][0][er];
        const float rsum = red[et][0][1][er] + red[et][1][1][er]
                         + red[et][2][1][er] + red[et][3][1][er];

        const size_t idx = (size_t)b * N_NODES
                         + (size_t)(grp * ROWS_PER_BLOCK + et * 16 + er);
        const float Vv = V[idx];
        const float Vs = Vv - V_shift[idx];
        const float mm = m[idx], hh = h[idx], nn = n[idx];

        const float am = 0.1f  * (Vs + 40.f) / (1.f - __expf(-(Vs + 40.f) * 0.1f));
        const float bm = 4.f   * __expf(-(Vs + 65.f) * (1.f / 18.f));
        const float ah = 0.07f * __expf(-(Vs + 65.f) * 0.05f);
        const float bh = 1.f   / (1.f + __expf(-(Vs + 35.f) * 0.1f));
        const float an = 0.01f * (Vs + 55.f) / (1.f - __expf(-(Vs + 55.f) * 0.1f));
        const float bn = 0.125f* __expf(-(Vs + 65.f) * (1.f / 80.f));

        const float dm = am * (1.f - mm) - bm * mm;
        const float dh = ah * (1.f - hh) - bh * hh;
        const float dn = an * (1.f - nn) - bn * nn;

        const float I_ax = dotV - rsum * Vv;
        const float I_Na = g_Na[idx] * mm * mm * mm * hh * (Vv - E_Na[idx]);
        const float I_K  = g_K[idx]  * nn * nn * nn * nn * (Vv - E_K[idx]);
        const float I_L  = g_leak[idx] * (Vv - E_leak[idx]);
        const float dV   = (I_inj[idx] + I_ax - I_Na - I_K - I_L) / Cm[idx];

        float4 o = make_float4(dV, dm, dh, dn);
        *(float4*)(out + idx * 4) = o;       // (B, N, 4) packed store
    }
}

extern "C" void kernel_launch(void* const* d_in, const int* in_sizes, int n_in,
                              void* d_out, int out_size, void* d_ws, size_t ws_size,
                              hipStream_t stream) {
    const float* V       = (const float*)d_in[0];
    const float* m       = (const float*)d_in[1];
    const float* h       = (const float*)d_in[2];
    const float* n       = (const float*)d_in[3];
    const float* I_inj   = (const float*)d_in[4];
    const float* adj     = (const float*)d_in[5];
    const float* g_Na    = (const float*)d_in[6];
    const float* g_K     = (const float*)d_in[7];
    const float* g_leak  = (const float*)d_in[8];
    const float* E_Na    = (const float*)d_in[9];
    const float* E_K     = (const float*)d_in[10];
    const float* E_leak  = (const float*)d_in[11];
    const float* Cm      = (const float*)d_in[12];
    const float* V_shift = (const float*)d_in[13];
    float* out = (float*)d_out;

    // 128 batches x 32 row-groups; 8 waves/block = 2 tiles x 4 K-quarters.
    dim3 grid(B_NEURONS * (N_NODES / ROWS_PER_BLOCK));
    dim3 block(256);
    hh_wmma_kernel<<<grid, block, 0, stream>>>(
        V, m, h, n, I_inj, adj, g_Na, g_K, g_leak,
        E_Na, E_K, E_leak, Cm, V_shift, out);
}